// FNORCLNTurbulence_38208029065322
// MI455X (gfx1250) — compile-verified
//
#include <hip/hip_runtime.h>

// ---------------------------------------------------------------------------
// FNO + Navier-Stokes closure for MI455X (gfx1250, wave32, WMMA).
//
// Roofline: dominated by ~400MB of HBM streaming (FD stencil + tau
// broadcast) => ~17us at 23.3 TB/s. u (100MB) is L2-resident (192MB L2) and
// re-read 7x by the stencil, so output streams use non-temporal stores to
// avoid evicting it. The FNO closure (65536 points) runs in exact fp32 with
// its GEMM-shaped parts on V_WMMA_F32_16X16X4_F32 (compile-time K/N => fully
// unrolled 8-step K-chain per 16x16 tile, no integer division). FFTs are
// partial factorized DFTs (mode truncation touches only 32 ky x 16 kx).
// ---------------------------------------------------------------------------

typedef __attribute__((ext_vector_type(2))) float v2f;
typedef __attribute__((ext_vector_type(4))) float v4f;
typedef __attribute__((ext_vector_type(8))) float v8f;

#define NXD   128
#define N2D   16384      // 128*128
#define HIDD  32

__device__ __forceinline__ float gelu_tanh(float v) {
  // jax.nn.gelu default (approximate=True)
  float t = tanhf(0.7978845608028654f * (v + 0.044715f * v * v * v));
  return 0.5f * v * (1.0f + t);
}

#define BUILD_TWIDDLE_TABLE()                                             \
  __shared__ float tcos[NXD], tsin[NXD];                                  \
  if (threadIdx.x < NXD) {                                                \
    float ang = 6.283185307179586f * (float)threadIdx.x * (1.0f/128.0f);  \
    float sn, cs;                                                         \
    sincosf(ang, &sn, &cs);                                               \
    tcos[threadIdx.x] = cs; tsin[threadIdx.x] = sn;                       \
  }                                                                       \
  __syncthreads();

// --- lift: u2d = u[:,:,:,:,64]; x = u2d @ lift_w + lift_b  (3 -> 32) ------
// x stored spatial-major: x[((b*128+h)*128+w)*32 + c]
__global__ void lift_kernel(const float* __restrict__ u,
                            const float* __restrict__ lw,
                            const float* __restrict__ lb,
                            float* __restrict__ xA) {
  int p  = blockIdx.x * blockDim.x + threadIdx.x;   // 0..65535 = b*16384+hw
  int b  = p >> 14;
  int hw = p & (N2D - 1);
  float u0 = u[((size_t)(b * 3 + 0) * N2D + hw) * NXD + 64];
  float u1 = u[((size_t)(b * 3 + 1) * N2D + hw) * NXD + 64];
  float u2 = u[((size_t)(b * 3 + 2) * N2D + hw) * NXD + 64];
  float* o = xA + (size_t)p * HIDD;
#pragma unroll
  for (int c = 0; c < HIDD; ++c)
    o[c] = u0 * lw[c] + u1 * lw[HIDD + c] + u2 * lw[2 * HIDD + c] + lb[c];
}

// --- forward DFT along W (rfft, only kx=0..15 needed) ---------------------
// F1[(((b*128+y)*16+kx)*32+c] (complex interleaved)
__global__ void dft_x_kernel(const float* __restrict__ x,
                             float* __restrict__ F1) {
  BUILD_TWIDDLE_TABLE();
  int e  = blockIdx.x * blockDim.x + threadIdx.x;   // 262144
  int c  = e & 31;
  int kx = (e >> 5) & 15;
  int y  = (e >> 9) & 127;
  int b  = e >> 16;
  const float* row = x + ((size_t)((b << 7) + y) << 7) * HIDD + c;
  float ar = 0.f, ai = 0.f;
  for (int xx = 0; xx < NXD; ++xx) {
    float v = row[(size_t)xx * HIDD];
    int   t = (kx * xx) & 127;
    ar += v * tcos[t];        // e^{-i theta}
    ai -= v * tsin[t];
  }
  F1[2 * (size_t)e]     = ar;
  F1[2 * (size_t)e + 1] = ai;
}

// --- forward DFT along H for the 32 kept rows (0..15 and 112..127) --------
// XF[(((b*32+kyl)*16+kx)*32+c] complex; kyl 16..31 <=> rows 112..127
__global__ void dft_y_kernel(const float* __restrict__ F1,
                             float* __restrict__ XF) {
  BUILD_TWIDDLE_TABLE();
  int e   = blockIdx.x * blockDim.x + threadIdx.x;  // 65536
  int c   = e & 31;
  int kx  = (e >> 5) & 15;
  int kyl = (e >> 9) & 31;
  int b   = e >> 14;
  int ky  = (kyl < 16) ? kyl : (kyl + 96);          // 112 + (kyl-16)
  size_t base = (size_t)b * 65536 + (size_t)kx * 32 + c;  // y=0
  float ar = 0.f, ai = 0.f;
  for (int y = 0; y < NXD; ++y) {
    size_t cid = base + (size_t)y * 512;
    float vr = F1[2 * cid], vi = F1[2 * cid + 1];
    int   t  = (ky * y) & 127;
    float ct = tcos[t], st = tsin[t];
    ar += vr * ct + vi * st;   // (vr+ivi)*(ct - i st)
    ai += vi * ct - vr * st;
  }
  XF[2 * (size_t)e]     = ar;
  XF[2 * (size_t)e + 1] = ai;
}

// --- complex channel mixing: out[b,o,ky,kx] = sum_i XF[b,i]*(wr+i wi) -----
// M=4 per mode -> too small for a 16x16 WMMA tile; VALU is fine (tiny op).
__global__ void mix_kernel(const float* __restrict__ XF,
                           const float* __restrict__ swr,
                           const float* __restrict__ swi,
                           float* __restrict__ OUTm, int l) {
  int e   = blockIdx.x * blockDim.x + threadIdx.x;  // 65536
  int o   = e & 31;
  int kx  = (e >> 5) & 15;
  int kyl = (e >> 9) & 15;
  int s   = (e >> 13) & 1;
  int b   = e >> 14;
  int ky32 = (s << 4) + kyl;
  size_t xbase = ((size_t)((b * 32 + ky32) * 16 + kx)) * 32;
  // sw[l][s][i][o][kyl][kx]
  size_t wbase = (size_t)(l * 2 + s) * 262144 + (size_t)o * 256 + kyl * 16 + kx;
  float ar = 0.f, ai = 0.f;
  for (int i = 0; i < HIDD; ++i) {
    size_t cid = xbase + i;
    float xr = XF[2 * cid], xi = XF[2 * cid + 1];
    float wr = swr[wbase + (size_t)i * 8192];
    float wi = swi[wbase + (size_t)i * 8192];
    ar += xr * wr - xi * wi;
    ai += xr * wi + xi * wr;
  }
  size_t ocid = xbase + o;
  OUTm[2 * ocid]     = ar;
  OUTm[2 * ocid + 1] = ai;
}

// --- inverse DFT along H (only 32 nonzero ky modes) -----------------------
__global__ void idft_y_kernel(const float* __restrict__ OUTm,
                              float* __restrict__ G) {
  BUILD_TWIDDLE_TABLE();
  int e  = blockIdx.x * blockDim.x + threadIdx.x;   // 262144
  int o  = e & 31;
  int kx = (e >> 5) & 15;
  int y  = (e >> 9) & 127;
  int b  = e >> 16;
  size_t base = (size_t)b * 16384 + (size_t)kx * 32 + o;  // kyl=0
  float ar = 0.f, ai = 0.f;
#pragma unroll
  for (int kyl = 0; kyl < 32; ++kyl) {
    int ky = (kyl < 16) ? kyl : (kyl + 96);
    int t  = (ky * y) & 127;
    float ct = tcos[t], st = tsin[t];
    size_t cid = base + (size_t)kyl * 512;
    float vr = OUTm[2 * cid], vi = OUTm[2 * cid + 1];
    ar += vr * ct - vi * st;   // (vr+ivi)*(ct + i st)
    ai += vr * st + vi * ct;
  }
  G[2 * (size_t)e]     = ar;
  G[2 * (size_t)e + 1] = ai;
}

// --- inverse rfft along W (kx=0..15 nonzero), real out, 1/(H*W) norm ------
__global__ void idft_x_kernel(const float* __restrict__ G,
                              float* __restrict__ xs) {
  BUILD_TWIDDLE_TABLE();
  int e  = blockIdx.x * blockDim.x + threadIdx.x;   // 2097152
  int o  = e & 31;
  int xx = (e >> 5) & 127;
  int y  = (e >> 12) & 127;
  int b  = e >> 19;
  size_t base = ((size_t)((b << 7) + y) * 16) * 32 + o;   // kx=0
  float acc = G[2 * base];                  // kx=0 counted once (real part)
#pragma unroll
  for (int kx = 1; kx < 16; ++kx) {
    int t = (kx * xx) & 127;
    size_t cid = base + (size_t)kx * 32;
    float gr = G[2 * cid], gi = G[2 * cid + 1];
    acc += 2.0f * (gr * tcos[t] - gi * tsin[t]);
  }
  xs[(size_t)e] = acc * (1.0f / 16384.0f);
}

// --- WMMA pointwise conv: Out = gelu(X @ W + bias [+ spec]) ---------------
// X spatial-major (M=65536 rows, K channels). V_WMMA_F32_16X16X4_F32,
// one 16x16 D-tile per wave; K, NOUT compile-time => shift/mask tile math
// and a fully unrolled 8x WMMA chain.
// Fragment layout per ISA 7.12.2 (f32): A lanes 0-15 hold {K=k0,k0+1} of
// row M=lane; lanes 16-31 hold {K=k0+2,k0+3}. B mirrored with N=lane%16.
// C/D: vgpr r = row (m0 + r + 8*(lane>=16)), col n0 + lane%16.
template <int K, int NOUT, bool ADDSPEC>
__global__ void pconv_wmma_kernel(const float* __restrict__ X,
                                  const float* __restrict__ W,
                                  const float* __restrict__ bias,
                                  const float* __restrict__ spec,
                                  float* __restrict__ Out) {
  constexpr int NTILES = NOUT >> 4;              // power of two (2 or 8)
  int wave = (blockIdx.x * blockDim.x + threadIdx.x) >> 5;
  int lane = threadIdx.x & 31;
  int tn = wave & (NTILES - 1);
  int tm = wave / NTILES;                         // constant power-of-two
  int l16  = lane & 15;
  int half = lane >> 4;
  int m0 = tm << 4;
  int n  = (tn << 4) + l16;

  v8f acc = {};
#pragma unroll
  for (int k0 = 0; k0 < K; k0 += 4) {
    int kb = k0 + 2 * half;
    v2f a = *(const v2f*)(X + (size_t)(m0 + l16) * K + kb);
    v2f bfrag;
    bfrag.x = W[(size_t)kb * NOUT + n];
    bfrag.y = W[(size_t)(kb + 1) * NOUT + n];
    acc = __builtin_amdgcn_wmma_f32_16x16x4_f32(
        /*neg_a=*/false, a, /*neg_b=*/false, bfrag,
        /*c_mod=*/(short)0, acc, /*reuse_a=*/false, /*reuse_b=*/false);
  }

  float bn = bias[n];
#pragma unroll
  for (int r = 0; r < 8; ++r) {
    int m = m0 + r + (half << 3);
    size_t idx = (size_t)m * NOUT + n;
    float v = acc[r] + bn;
    if (ADDSPEC) v += spec[idx];
    Out[idx] = gelu_tanh(v);
  }
}

// --- fc2: tau2d[b,o,h,w] = Y @ fc2_w + fc2_b  (128 -> 6) ------------------
__global__ void fc2_kernel(const float* __restrict__ Y,
                           const float* __restrict__ w,
                           const float* __restrict__ bb,
                           float* __restrict__ tau2d) {
  int p  = blockIdx.x * blockDim.x + threadIdx.x;   // 65536
  int b  = p >> 14;
  int hw = p & (N2D - 1);
  const float* yp = Y + (size_t)p * 128;
  float acc[6];
#pragma unroll
  for (int o = 0; o < 6; ++o) acc[o] = bb[o];
  for (int i = 0; i < 128; ++i) {
    float v = yp[i];
#pragma unroll
    for (int o = 0; o < 6; ++o) acc[o] += v * w[i * 6 + o];
  }
#pragma unroll
  for (int o = 0; o < 6; ++o)
    tau2d[((size_t)(b * 6 + o) << 14) + hw] = acc[o];
}

// --- fused FD stencil + 0.01*tau: du_dt half of d_out ---------------------
// u fits in the 192MB L2; stencil reads hit L2. Output stores use the
// non-temporal hint (TH=NT) so the 300MB of streaming writes don't evict u.
__global__ void fd_kernel(const float* __restrict__ u,
                          const float* __restrict__ tau2d,
                          float* __restrict__ out) {
  long long t = (long long)blockIdx.x * blockDim.x + threadIdx.x;  // 25165824
  int i4 = (int)(t & 127);
  int i3 = (int)((t >> 7) & 127);
  int i2 = (int)((t >> 14) & 127);
  int bc = (int)(t >> 21);            // b*3+c
  int c  = bc % 3;
  int b  = bc / 3;
  const float* U = u + ((size_t)bc << 21);
  int r23 = (i2 << 14) | (i3 << 7);
  float uc  = U[r23 | i4];
  float uxp = U[(((i2 + 1) & 127) << 14) | (i3 << 7) | i4];
  float uxm = U[(((i2 + 127) & 127) << 14) | (i3 << 7) | i4];
  float uyp = U[(i2 << 14) | (((i3 + 1) & 127) << 7) | i4];
  float uym = U[(i2 << 14) | (((i3 + 127) & 127) << 7) | i4];
  float uzp = U[r23 | ((i4 + 1) & 127)];
  float uzm = U[r23 | ((i4 + 127) & 127)];
  const float dxv    = 6.283185307179586f / 128.0f;
  const float inv2dx = 1.0f / (2.0f * dxv);
  const float invdx2 = 1.0f / (dxv * dxv);
  float grad = (c == 0) ? (uxp - uxm) : (c == 1) ? (uyp - uym) : (uzp - uzm);
  float conv = -uc * grad * inv2dx;
  float lap  = (uxp + uxm + uyp + uym + uzp + uzm - 6.0f * uc) * invdx2;
  float tau  = tau2d[((size_t)(b * 6 + c) << 14) + (i2 << 7) + i3];
  __builtin_nontemporal_store(conv + 0.002f * lap + 0.01f * tau, out + t);
}

// --- tau broadcast over z: streaming non-temporal float4 writes -----------
__global__ void bcast_kernel(const float* __restrict__ tau2d,
                             float* __restrict__ out) {
  long long t = (long long)blockIdx.x * blockDim.x + threadIdx.x;  // 12582912
  int    zq  = (int)(t & 31);
  size_t row = (size_t)(t >> 5);
  float  v   = tau2d[row];
  v4f f4 = {v, v, v, v};
  __builtin_nontemporal_store(f4, (v4f*)(out + row * 128 + zq * 4));
}

extern "C" void kernel_launch(void* const* d_in, const int* in_sizes, int n_in,
                              void* d_out, int out_size, void* d_ws, size_t ws_size,
                              hipStream_t stream) {
  const float* u      = (const float*)d_in[0];
  const float* lift_w = (const float*)d_in[1];
  const float* lift_b = (const float*)d_in[2];
  const float* sw_r   = (const float*)d_in[3];
  const float* sw_i   = (const float*)d_in[4];
  const float* pw_w   = (const float*)d_in[5];
  const float* pw_b   = (const float*)d_in[6];
  const float* fc1_w  = (const float*)d_in[7];
  const float* fc1_b  = (const float*)d_in[8];
  const float* fc2_w  = (const float*)d_in[9];
  const float* fc2_b  = (const float*)d_in[10];
  float* out = (float*)d_out;

  // Workspace layout (floats); total = 16,384,000 floats (~65.5 MB).
  float* ws   = (float*)d_ws;
  float* xA   = ws;                   // 2097152  (B,H,W,32) activations
  float* xB   = xA   + 2097152;       // 2097152  ping-pong
  float* F1   = xB   + 2097152;       //  524288  (B,128,16,32) complex
  float* G    = F1   +  524288;       //  524288  (B,128,16,32) complex
  float* XF   = G    +  524288;       //  131072  (B,32,16,32)  complex
  float* OUTm = XF   +  131072;       //  131072  (B,32,16,32)  complex
  float* xs   = OUTm +  131072;       // 2097152  spectral branch output
  float* Y    = xs   + 2097152;       // 8388608  (B,H,W,128) fc1 output
  float* tau2 = Y    + 8388608;       //  393216  (B,6,H,W)

  lift_kernel<<<256, 256, 0, stream>>>(u, lift_w, lift_b, xA);

  float* cur = xA;
  float* nxt = xB;
  for (int l = 0; l < 4; ++l) {
    dft_x_kernel <<<1024, 256, 0, stream>>>(cur, F1);
    dft_y_kernel <<< 256, 256, 0, stream>>>(F1, XF);
    mix_kernel   <<< 256, 256, 0, stream>>>(XF, sw_r, sw_i, OUTm, l);
    idft_y_kernel<<<1024, 256, 0, stream>>>(OUTm, G);
    idft_x_kernel<<<8192, 256, 0, stream>>>(G, xs);
    // fused: nxt = gelu(cur @ pw_w[l] + pw_b[l] + xs)   [WMMA]
    pconv_wmma_kernel<HIDD, HIDD, true><<<1024, 256, 0, stream>>>(
        cur, pw_w + l * HIDD * HIDD, pw_b + l * HIDD, xs, nxt);
    float* tmp = cur; cur = nxt; nxt = tmp;
  }

  // fc1: Y = gelu(cur @ fc1_w + fc1_b)   [WMMA, N=128]
  pconv_wmma_kernel<HIDD, 128, false><<<4096, 256, 0, stream>>>(
      cur, fc1_w, fc1_b, nullptr, Y);

  fc2_kernel<<<256, 256, 0, stream>>>(Y, fc2_w, fc2_b, tau2);

  // d_out = [du_dt (25165824) | tau (50331648)]
  fd_kernel   <<<98304, 256, 0, stream>>>(u, tau2, out);
  bcast_kernel<<<49152, 256, 0, stream>>>(tau2, out + 25165824);
}